// Attention_50379966382361
// MI455X (gfx1250) — compile-verified
//
#include <hip/hip_runtime.h>
#include <hip/hip_bf16.h>
#include <math.h>

// ---------------------------------------------------------------------------
// Factorized spatial attention, fp32, CDNA5 (gfx1250).
//   B=2, F=16, N_SP=784, VIS=512, DIM=512, HEADS=8, HD=64
// Pipeline: QKV GEMM -> per-(b,f,h) flash attention -> proj GEMM (+bias).
// All matrix math uses V_WMMA_F32_16X16X4_F32 (full fp32, matches reference).
// GEMM B-tiles are staged to LDS with GLOBAL_LOAD_ASYNC_TO_LDS_B128 (ASYNCcnt),
// double-buffered against the WMMA stream.
// ---------------------------------------------------------------------------

typedef __attribute__((ext_vector_type(2))) float v2f;
typedef __attribute__((ext_vector_type(8))) float v8f;

#define B_SZ   2
#define F_SZ   16
#define NSP    784
#define VIS_D  512
#define D_MOD  512
#define NHEADS 8
#define HDIM   64
#define QTILES 49           // NSP / 16
#define QKVW   (3 * D_MOD)  // 1536 floats per qkv row

static __device__ __forceinline__ v8f vzero8() {
  v8f r;
#pragma unroll
  for (int i = 0; i < 8; ++i) r[i] = 0.0f;
  return r;
}

// D = A(16x4 f32) * B(4x16 f32) + C(16x16 f32)
static __device__ __forceinline__ v8f wmma4_f32(v2f a, v2f b, v8f c) {
  return __builtin_amdgcn_wmma_f32_16x16x4_f32(false, a, false, b, (short)0, c,
                                               false, false);
}

// Async 16-byte global -> LDS copy (per-lane addresses), tracked by ASYNCcnt.
static __device__ __forceinline__ void async_g2l_b128(unsigned lds_off,
                                                      const float* gaddr) {
  asm volatile("global_load_async_to_lds_b128 %0, %1, off"
               :
               : "v"(lds_off), "v"(gaddr)
               : "memory");
}

// ---------------------------------------------------------------------------
// C[M,N] = A[M,K] * Bw[N,K]^T (+ bias[N])
// Block: 256 threads = 8 waves; each wave computes a 16(M) x 64(N) strip.
// The 64xKC B tile is shared by all 8 waves -> staged once per block in LDS
// via async copies, double-buffered.
// fp32 WMMA operand layouts (ISA 7.12.2):
//   A 16x4 : lane 0-15 -> M=lane, K = 2*(lane>=16) + reg  => float2 @ (row, k0)
//   B 4x16 : lane -> N=lane%16, K = 2*(lane>=16) + reg    => float2 @ (col-row, k0)
//   C 16x16: reg i -> rows i (lanes 0-15) and i+8 (lanes 16-31), lane%16 -> N
// ---------------------------------------------------------------------------
template <int KTOT, bool HAS_BIAS>
__global__ __launch_bounds__(256) void gemm_nt_wmma(
    const float* __restrict__ A, const float* __restrict__ Bw,
    const float* __restrict__ bias, float* __restrict__ C, int N) {
  constexpr int KC   = 32;       // K depth per stage
  constexpr int BROW = KC + 4;   // pad: b128 dests stay 16B aligned (36*4=144),
                                 // and 36*r mod 64 distinct for r=0..15 (banks)
  __shared__ float bsm[2][64 * BROW];

  const int tid  = threadIdx.x;
  const int lane = tid & 31;
  const int wave = tid >> 5;
  const int lm   = lane & 15;
  const int lh   = lane >> 4;  // 0: K pair {0,1}, 1: K pair {2,3}
  const int tileM = blockIdx.x * 128 + wave * 16;
  const int tileN = blockIdx.y * 64;

  v8f acc[4];
#pragma unroll
  for (int t = 0; t < 4; ++t) acc[t] = vzero8();

  const float* arow = A + (size_t)(tileM + lm) * KTOT;

  // --- async stage of one 64xKC B tile into bsm[buf] (2 x b128 per thread) --
  auto stage_b = [&](int sbuf, int kk) {
#pragma unroll
    for (int r = 0; r < 2; ++r) {
      const int e    = tid + r * 256;   // 512 chunks of 4 floats
      const int row  = e >> 3;          // 8 chunks per 32-float row
      const int kc4  = (e & 7) << 2;
      const unsigned lds_off =
          (unsigned)(uintptr_t)(&bsm[sbuf][row * BROW + kc4]);
      const float* g = Bw + (size_t)(tileN + row) * KTOT + kk + kc4;
      async_g2l_b128(lds_off, g);
    }
  };

  int buf = 0;
  stage_b(0, 0);
  for (int kk = 0; kk < KTOT; kk += KC) {
    if (kk + KC < KTOT) {
      stage_b(buf ^ 1, kk + KC);
      // our 2 newest copies may stay in flight; the 2 for this tile are done
      asm volatile("s_wait_asynccnt 0x2" ::: "memory");
    } else {
      asm volatile("s_wait_asynccnt 0x0" ::: "memory");
    }
    __syncthreads();  // tile visible to all 8 waves

    const float* bbase = &bsm[buf][0];
#pragma unroll
    for (int k4 = 0; k4 < KC; k4 += 4) {
      const int k0 = k4 + 2 * lh;
      const v2f af = *(const v2f*)(arow + kk + k0);
#pragma unroll
      for (int t = 0; t < 4; ++t) {
        const v2f bf = *(const v2f*)(bbase + (t * 16 + lm) * BROW + k0);
        acc[t] = wmma4_f32(af, bf, acc[t]);
      }
    }
    __syncthreads();  // done reading before next stage overwrites this buffer
    buf ^= 1;
  }

#pragma unroll
  for (int t = 0; t < 4; ++t) {
    const int n = tileN + t * 16 + lm;
    const float bv = HAS_BIAS ? bias[n] : 0.0f;
#pragma unroll
    for (int i = 0; i < 8; ++i) {
      const int m = tileM + i + 8 * lh;
      C[(size_t)m * N + n] = acc[t][i] + bv;
    }
  }
}

// ---------------------------------------------------------------------------
// Flash-attention over the spatial axis. One wave (32 threads) per block,
// block = (b, f, head, query-tile-of-16). Streams 49 key/value tiles with
// online softmax; P is re-laid-out C->A through a 1KB LDS staging tile.
// qkv rows: [B*F*NSP, 1536] = (q | k | v), head h occupies cols h*64..h*64+63.
// out rows: [B*F*NSP, 512].
// ---------------------------------------------------------------------------
__global__ __launch_bounds__(32) void spatial_attn_wmma(
    const float* __restrict__ qkv, float* __restrict__ out) {
  __shared__ float pbuf[16][17];  // +1 pad: kill 2-way LDS bank conflicts

  const int lane = threadIdx.x & 31;
  const int lm   = lane & 15;
  const int lh   = lane >> 4;

  int blk = blockIdx.x;
  const int qt = blk % QTILES; blk /= QTILES;
  const int h  = blk % NHEADS; blk /= NHEADS;
  const int f  = blk % F_SZ;   blk /= F_SZ;
  const int b  = blk;

  const size_t row_base = (size_t)(b * F_SZ + f) * NSP;  // first token of frame
  const int QOFF = h * HDIM;
  const int KOFF = D_MOD + h * HDIM;
  const int VOFF = 2 * D_MOD + h * HDIM;
  const float scale = 0.125f;  // 64^-0.5

  // Q fragments: 16 chunks of K=4 (A-operand layout), pre-scaled.
  v2f qf[16];
  {
    const float* qrow = qkv + (row_base + qt * 16 + lm) * (size_t)QKVW + QOFF;
#pragma unroll
    for (int c = 0; c < 16; ++c) {
      v2f t = *(const v2f*)(qrow + 4 * c + 2 * lh);
      qf[c][0] = t[0] * scale;
      qf[c][1] = t[1] * scale;
    }
  }

  v8f o[4];  // O accumulator, 16 rows x 64 cols (4 col tiles)
#pragma unroll
  for (int t = 0; t < 4; ++t) o[t] = vzero8();
  float mrun[8], lrun[8];  // per-row running max / sum (row = i + 8*lh)
#pragma unroll
  for (int i = 0; i < 8; ++i) { mrun[i] = -1e30f; lrun[i] = 0.0f; }

  for (int j = 0; j < QTILES; ++j) {
    // ---- S = Q * K_j^T : 16 fp32 WMMAs -------------------------------------
    v8f s = vzero8();
    const float* krow = qkv + (row_base + j * 16 + lm) * (size_t)QKVW + KOFF;
#pragma unroll
    for (int c = 0; c < 16; ++c) {
      const v2f kf = *(const v2f*)(krow + 4 * c + 2 * lh);
      s = wmma4_f32(qf[c], kf, s);
    }

    // ---- online softmax (rows live in 16-lane halves) ----------------------
    float alpha[8];
#pragma unroll
    for (int i = 0; i < 8; ++i) {
      float v = s[i];
#pragma unroll
      for (int d = 1; d < 16; d <<= 1) v = fmaxf(v, __shfl_xor(v, d, 32));
      const float mnew = fmaxf(mrun[i], v);
      alpha[i] = __expf(mrun[i] - mnew);
      mrun[i] = mnew;
      const float p = __expf(s[i] - mnew);
      s[i] = p;
      float rs = p;
#pragma unroll
      for (int d = 1; d < 16; d <<= 1) rs += __shfl_xor(rs, d, 32);
      lrun[i] = lrun[i] * alpha[i] + rs;
    }
#pragma unroll
    for (int t = 0; t < 4; ++t)
#pragma unroll
      for (int i = 0; i < 8; ++i) o[t][i] *= alpha[i];

    // ---- C-layout -> A-layout for P via LDS --------------------------------
    __syncthreads();
#pragma unroll
    for (int i = 0; i < 8; ++i) pbuf[i + 8 * lh][lm] = s[i];
    __syncthreads();
    v2f pf[4];
#pragma unroll
    for (int c = 0; c < 4; ++c) {
      const int u0 = 4 * c + 2 * lh;
      pf[c][0] = pbuf[lm][u0 + 0];
      pf[c][1] = pbuf[lm][u0 + 1];
    }

    // ---- O += P * V_j : 16 fp32 WMMAs --------------------------------------
    const float* vbase = qkv + (row_base + j * 16) * (size_t)QKVW + VOFF;
#pragma unroll
    for (int c = 0; c < 4; ++c) {
      const int u0 = 4 * c + 2 * lh;
#pragma unroll
      for (int t = 0; t < 4; ++t) {
        v2f vf;
        vf[0] = vbase[(size_t)(u0 + 0) * QKVW + t * 16 + lm];
        vf[1] = vbase[(size_t)(u0 + 1) * QKVW + t * 16 + lm];
        o[t] = wmma4_f32(pf[c], vf, o[t]);
      }
    }
  }

  // ---- normalize and write -------------------------------------------------
  float inv[8];
#pragma unroll
  for (int i = 0; i < 8; ++i) inv[i] = 1.0f / lrun[i];
  float* obase = out + (row_base + qt * 16) * (size_t)D_MOD + h * HDIM;
#pragma unroll
  for (int t = 0; t < 4; ++t)
#pragma unroll
    for (int i = 0; i < 8; ++i)
      obase[(size_t)(i + 8 * lh) * D_MOD + t * 16 + lm] = o[t][i] * inv[i];
}

// ---------------------------------------------------------------------------
extern "C" void kernel_launch(void* const* d_in, const int* in_sizes, int n_in,
                              void* d_out, int out_size, void* d_ws,
                              size_t ws_size, hipStream_t stream) {
  (void)in_sizes; (void)n_in; (void)out_size; (void)ws_size;
  const float* x     = (const float*)d_in[0];  // [2, 12544, 512]
  const float* Wqkv  = (const float*)d_in[1];  // [1536, 512]
  const float* Wproj = (const float*)d_in[2];  // [512, 512]
  const float* bproj = (const float*)d_in[3];  // [512]
  float* outp = (float*)d_out;

  const int M = B_SZ * F_SZ * NSP;  // 25088 rows (multiple of 128)

  float* qkv  = (float*)d_ws;                 // M * 1536 floats (154 MB)
  float* attn = qkv + (size_t)M * QKVW;       // M * 512 floats  (51 MB)

  // 1) qkv = x @ Wqkv^T
  dim3 g1(M / 128, QKVW / 64);
  gemm_nt_wmma<VIS_D, false><<<g1, 256, 0, stream>>>(x, Wqkv, nullptr, qkv,
                                                     QKVW);

  // 2) per-(b,f,head) spatial flash attention
  spatial_attn_wmma<<<B_SZ * F_SZ * NHEADS * QTILES, 32, 0, stream>>>(qkv,
                                                                      attn);

  // 3) out = attn @ Wproj^T + bproj
  dim3 g2(M / 128, D_MOD / 64);
  gemm_nt_wmma<D_MOD, true><<<g2, 256, 0, stream>>>(attn, Wproj, bproj, outp,
                                                    D_MOD);
}